// QueryAndGroup_19121194402076
// MI455X (gfx1250) — compile-verified
//
#include <hip/hip_runtime.h>

typedef __attribute__((ext_vector_type(2))) float v2f;
typedef __attribute__((ext_vector_type(8))) float v8f;

// Problem constants (from reference setup_inputs)
constexpr int B = 4;
constexpr int N = 16384;
constexpr int P = 2048;
constexpr int C = 64;
constexpr int K = 32;
constexpr float R2 = 0.01f;             // RADIUS^2
constexpr float FINF = 3.0e38f;

constexpr int QT    = 16;               // queries per block (one WMMA M-tile)
constexpr int CHUNK = 512;              // points staged per iteration
constexpr int LDW   = CHUNK + 4;        // LDS row stride: 16B-aligned rows, bank-spread
constexpr int NW    = 4;                // waves per block
constexpr int NT    = NW * 32;          // 128 threads
constexpr int SLICES  = NT / QT;        // 8 scan slices per query
constexpr int SLICE_W = CHUNK / SLICES; // 64 points per slice per chunk

__global__ __launch_bounds__(NT)
void qg_knn_wmma_kernel(const float* __restrict__ xyz,
                        const float* __restrict__ nxyz,
                        const float* __restrict__ feat,
                        float* __restrict__ out) {
    __shared__ float d2[QT * LDW];      // 33,024 B distance tiles
    __shared__ float lst_d[NT * K];     // 16 KB per-thread sorted top-32 dists
    __shared__ int   lst_i[NT * K];     // 16 KB per-thread sorted top-32 idx
    __shared__ float ld_q[QT][4];       // qx,qy,qz,|q|^2
    __shared__ float md[QT * K];        // merged sorted dists
    __shared__ int   fi[QT * K];        // final masked neighbor indices

    const int t    = threadIdx.x;       // 0..127
    const int lane = t & 31;            // wave32 lane
    const int wid  = t >> 5;            // wave id 0..3
    const int half = lane >> 4;         // WMMA lane half
    const int l16  = lane & 15;

    const int blk   = blockIdx.x;       // 0..511
    const int b     = blk / (P / QT);
    const int ptile = blk - b * (P / QT);
    const int p0    = ptile * QT;

    // ---- stage 16 query points; init private top-K lists to +INF ----
    if (t < QT) {
        const float* qp = nxyz + ((size_t)b * P + p0 + t) * 3;
        float qx = qp[0], qy = qp[1], qz = qp[2];
        ld_q[t][0] = qx; ld_q[t][1] = qy; ld_q[t][2] = qz;
        ld_q[t][3] = qx * qx + qy * qy + qz * qz;
    }
    const int lbase = t * K;
#pragma unroll
    for (int k = 0; k < K; ++k) lst_d[lbase + k] = FINF;
    __syncthreads();

    // ---- WMMA A operand (16x4 f32): lanes 0-15 hold K=0,1; lanes 16-31 K=2,3
    // A[m] = (qx, qy, qz, 1)
    v2f Av;
    Av[0] = half ? ld_q[l16][2] : ld_q[l16][0];
    Av[1] = half ? 1.0f         : ld_q[l16][1];
    // C operand (16x16 f32 C/D layout): VGPR r -> row r + 8*half ; C = |q_row|^2
    v8f Cv;
#pragma unroll
    for (int r = 0; r < 8; ++r) Cv[r] = ld_q[r + 8 * half][3];

    const int qm = t & (QT - 1);        // owned query row for scan phase
    const int sl = t >> 4;              // slice 0..7
    float worst = FINF;                 // register-resident K-th smallest

    for (int cb = 0; cb < N; cb += CHUNK) {
        // --- distance tiles via V_WMMA_F32_16X16X4_F32 ---
        for (int tt = wid; tt < CHUNK / 16; tt += NW) {
            const int n = cb + tt * 16 + l16;
            const float* pp = xyz + ((size_t)b * N + n) * 3;
            float px = pp[0], py = pp[1], pz = pp[2];
            // B operand (4x16 f32): lanes 0-15 rows K=0,1; lanes 16-31 rows K=2,3
            // B[n] = (-2px, -2py, -2pz, |p|^2)   (ternaries -> v_cndmask, no branches)
            v2f Bv;
            Bv[0] = half ? (-2.0f * pz) : (-2.0f * px);
            Bv[1] = half ? (px * px + py * py + pz * pz) : (-2.0f * py);
            v8f d = __builtin_amdgcn_wmma_f32_16x16x4_f32(
                false, Av, false, Bv, (short)0, Cv, false, false);
            const int col  = tt * 16 + l16;
            const int rowb = 8 * half;
#pragma unroll
            for (int r = 0; r < 8; ++r)
                d2[(rowb + r) * LDW + col] = d[r];
        }
        __syncthreads();

        // --- scan: stream 64 dists into LDS-resident sorted top-32 ---
        const float* rowp = &d2[qm * LDW + sl * SLICE_W];
        const int nb = cb + sl * SLICE_W;
#pragma clang loop unroll(disable)
        for (int jj = 0; jj < SLICE_W; jj += 4) {
            float4 dv = *reinterpret_cast<const float4*>(rowp + jj);
#pragma unroll
            for (int e = 0; e < 4; ++e) {
                float d = (e == 0) ? dv.x : (e == 1) ? dv.y : (e == 2) ? dv.z : dv.w;
                if (d < worst) {                       // rare after warm-up
                    int pos = K - 1;
#pragma clang loop unroll(disable)
                    while (pos > 0) {
                        float prev = lst_d[lbase + pos - 1];
                        if (prev <= d) break;
                        lst_d[lbase + pos] = prev;
                        lst_i[lbase + pos] = lst_i[lbase + pos - 1];
                        --pos;
                    }
                    lst_d[lbase + pos] = d;
                    lst_i[lbase + pos] = nb + jj + e;
                    worst = lst_d[lbase + K - 1];
                }
            }
        }
        __syncthreads();
    }

    // ---- 8-way merge of sorted lists + radius mask (one thread per query) ----
    if (t < QT) {
        int h[SLICES];
#pragma unroll
        for (int s = 0; s < SLICES; ++s) h[s] = 0;
#pragma clang loop unroll(disable)
        for (int k = 0; k < K; ++k) {
            float best = FINF; int bs = 0;
#pragma unroll
            for (int s = 0; s < SLICES; ++s) {
                int hp = h[s];
                float dv = (hp < K) ? lst_d[(t + QT * s) * K + hp] : FINF;
                if (dv < best) { best = dv; bs = s; }
            }
            md[t * K + k] = best;
            fi[t * K + k] = lst_i[(t + QT * bs) * K + h[bs]];
            h[bs]++;
        }
        int inball = 0;
#pragma clang loop unroll(disable)
        for (int k = 0; k < K; ++k) inball += (md[t * K + k] <= R2) ? 1 : 0;
        const int lastp = (inball > 0) ? (inball - 1) : 0;
        const int li    = fi[t * K + lastp];
#pragma clang loop unroll(disable)
        for (int k = 0; k < K; ++k)
            if (k >= inball) fi[t * K + k] = li;
    }
    __syncthreads();

    // ---- output: (B, 3+C, P, K); one wave per (p, channel) row, lane = k ----
    for (int row = wid; row < QT * (C + 3); row += NW) {
        const int pl = row / (C + 3);
        const int c  = row - pl * (C + 3);
        const int p  = p0 + pl;
        const int nk = fi[pl * K + lane];
        float v;
        if (c < 3) v = xyz[((size_t)b * N + nk) * 3 + c] - ld_q[pl][c];
        else       v = feat[((size_t)b * C + (c - 3)) * N + nk];
        out[(((size_t)b * (C + 3) + c) * P + p) * K + lane] = v;
    }
}

extern "C" void kernel_launch(void* const* d_in, const int* in_sizes, int n_in,
                              void* d_out, int out_size, void* d_ws, size_t ws_size,
                              hipStream_t stream) {
    (void)in_sizes; (void)n_in; (void)out_size; (void)d_ws; (void)ws_size;
    const float* xyz  = (const float*)d_in[0];
    const float* nxyz = (const float*)d_in[1];
    const float* feat = (const float*)d_in[2];
    float* out = (float*)d_out;
    dim3 grid(B * (P / QT));   // 512 blocks
    qg_knn_wmma_kernel<<<grid, NT, 0, stream>>>(xyz, nxyz, feat, out);
}